// DemoRNN_37520834298127
// MI455X (gfx1250) — compile-verified
//
#include <hip/hip_runtime.h>

// CDNA5 (gfx1250) wave32 WMMA RNN.
//   H_new^T = tanh( W_hh * H^T + pre^T ),  pre = x*W_ih + b_ih + b_hh
// A (SRC0) = W_hh with PERMUTED columns (16x12, zero padded), B (SRC1) = H^T,
// C/D = fp32 16x16 accumulator (batch on lanes, hidden on VGPRs).
// Column permutation CU makes the D->B relayout need only 2 ds_swizzles/step.

typedef __attribute__((ext_vector_type(2))) float v2f;
typedef __attribute__((ext_vector_type(8))) float v8f;

#define HID 10

// Column unit order: B slot q holds hidden unit CU[q].
//  low-lane slots {0,1,4,5,8,9} <- units {0,1,2,3,4,5} (stay in low half)
//  high-lane slots {2,3}        <- units {6,7}         (cross-half: swizzle)
//  high-lane slots {6,7}        <- units {8,9}         (stay in high half)
//  slots {10,11} are zero columns of A (don't-care B rows)
__device__ __constant__ const int CU_[12] = {0, 1, 6, 7, 2, 3, 8, 9, 4, 5, 10, 11};

// swap values between lane halves (lane ^ 16) of a wave32: ds_swizzle SWAPX16
__device__ __forceinline__ float swap16(float v) {
  return __int_as_float(__builtin_amdgcn_ds_swizzle(__float_as_int(v), 0x401F));
}

__device__ __forceinline__ float tanh_fast(float v) {
#if __has_builtin(__builtin_amdgcn_tanhf)
  return __builtin_amdgcn_tanhf(v);          // v_tanh_f32 (gfx1250 trans op)
#else
  float e = __builtin_amdgcn_exp2f(v * 2.8853900817779268f);  // e^(2v)
  return 1.0f - 2.0f / (e + 1.0f);
#endif
}

template <bool V> struct BoolT { static constexpr bool value = V; };

__global__ __launch_bounds__(32) void rnn_wmma_kernel(
    const float* __restrict__ x,       // (B, T)
    const float* __restrict__ hidden,  // (B, HID)
    const float* __restrict__ W_ih,    // (HID)
    const float* __restrict__ W_hh,    // (HID, HID)
    const float* __restrict__ b_ih,    // (HID)
    const float* __restrict__ b_hh,    // (HID)
    const float* __restrict__ W_out,   // (HID)
    const float* __restrict__ b_out,   // (1)
    float* __restrict__ out,           // [0,T): batch-0 proj; [T, T+B*HID): h_last
    int B, int T) {
  const int  lane  = threadIdx.x;
  const int  half  = lane >> 4;     // 0 = low lanes, 1 = high lanes
  const int  nl    = lane & 15;
  const bool isLow = (half == 0);
  const int  base  = blockIdx.x * 16;
  const int  batchRow = base + nl;
  const int  xrow  = (batchRow < B) ? batchRow : (B - 1);
  const size_t rowoff = (size_t)xrow * (size_t)T;

  // ---- A[p][q] = W_hh[p][CU[q]], zero-padded. f32 16x4 A layout (ISA 7.12.2):
  // VGPR0: lanes0-15 K=4c, lanes16-31 K=4c+2 (M = lane&15); VGPR1: K+1.
  v2f Am[3];
#pragma unroll
  for (int c = 0; c < 3; ++c) {
    int q = 4 * c + 2 * half;
    int u0 = CU_[q], u1 = CU_[q + 1];
    Am[c][0] = (nl < HID && u0 < HID) ? W_hh[nl * HID + u0] : 0.0f;
    Am[c][1] = (nl < HID && u1 < HID) ? W_hh[nl * HID + u1] : 0.0f;
  }

  // ---- per-lane constants: C/D VGPR v on this lane is hidden unit v+8*half
  float wihc[8], biasc[8], woutc[8];
#pragma unroll
  for (int v = 0; v < 8; ++v) {
    int h = v + 8 * half;
    bool ok = (h < HID);
    wihc[v]  = ok ? W_ih[h] : 0.0f;
    biasc[v] = ok ? (b_ih[h] + b_hh[h]) : 0.0f;
    woutc[v] = ok ? W_out[h] : 0.0f;
  }
  const float bout0 = b_out[0];

  // ---- B = H^T initial state, permuted columns
  v2f Bm[3];
#pragma unroll
  for (int c = 0; c < 3; ++c)
#pragma unroll
    for (int r = 0; r < 2; ++r) {
      int u = CU_[4 * c + 2 * half + r];
      Bm[c][r] = (u < HID && batchRow < B)
                     ? hidden[(size_t)batchRow * HID + u] : 0.0f;
    }

  float hv[8];
#pragma unroll
  for (int v = 0; v < 8; ++v) hv[v] = 0.0f;

  float xcur[16], xnxt[16];
#pragma unroll
  for (int q = 0; q < 16; ++q) xnxt[q] = 0.0f;

  auto loadChunk = [&](float* dst, int t0) {
    const float4* p = reinterpret_cast<const float4*>(x + rowoff + t0);
#pragma unroll
    for (int q = 0; q < 4; ++q) {
      float4 w = p[q];
      dst[4 * q + 0] = w.x; dst[4 * q + 1] = w.y;
      dst[4 * q + 2] = w.z; dst[4 * q + 3] = w.w;
    }
  };

  const int Tmain = T & ~15;

  auto body = [&](auto outc) {
    constexpr bool OUT = decltype(outc)::value;

    auto do_step = [&](float xv, int t) {
      // C = pre^T (rank-1): packed FMAs
      v8f acc;
#pragma unroll
      for (int v = 0; v < 8; ++v) acc[v] = fmaf(xv, wihc[v], biasc[v]);
      // Accumulate K slabs; chunk 0 (the one that needed swizzled B) last,
      // so its ds_swizzle latency hides under the other two WMMAs.
      acc = __builtin_amdgcn_wmma_f32_16x16x4_f32(false, Am[1], false, Bm[1],
                                                  (short)0, acc, false, false);
      acc = __builtin_amdgcn_wmma_f32_16x16x4_f32(false, Am[2], false, Bm[2],
                                                  (short)0, acc, false, false);
      acc = __builtin_amdgcn_wmma_f32_16x16x4_f32(false, Am[0], false, Bm[0],
                                                  (short)0, acc, false, false);
      // tanh units 6,7 first: their cross-half swizzles are the only DS ops.
      hv[6] = tanh_fast(acc[6]);
      hv[7] = tanh_fast(acc[7]);
      float s6 = swap16(hv[6]);
      float s7 = swap16(hv[7]);
#pragma unroll
      for (int v = 0; v < 6; ++v) hv[v] = tanh_fast(acc[v]);
      // Rebuild B (slot q holds unit CU[q]); only slots 2,3 cross halves.
      Bm[0][0] = isLow ? hv[0] : s6;     // q0:u0 | q2:u6
      Bm[0][1] = isLow ? hv[1] : s7;     // q1:u1 | q3:u7
      Bm[1][0] = isLow ? hv[2] : hv[0];  // q4:u2 | q6:u8 (in place)
      Bm[1][1] = isLow ? hv[3] : hv[1];  // q5:u3 | q7:u9 (in place)
      Bm[2][0] = hv[4];                  // q8:u4 | q10: don't care (A col = 0)
      Bm[2][1] = hv[5];                  // q9:u5 | q11: don't care
      if constexpr (OUT) {  // batch 0 tile: project + store
        float s = 0.0f;
#pragma unroll
        for (int v = 0; v < 8; ++v) s = fmaf(hv[v], woutc[v], s);
        s += swap16(s);                  // lane0 += lane16 partial
        if (lane == 0) out[t] = s + bout0;
      }
    };

    if (Tmain > 0) loadChunk(xcur, 0);
    for (int tc = 0; tc < Tmain; tc += 16) {
      // Warm L2/L0 four chunks ahead so the b128 clause below hits cache.
      if (tc + 64 < T) __builtin_prefetch(x + rowoff + tc + 64, 0, 3);
      int tn = tc + 16;
      if (tn < Tmain) loadChunk(xnxt, tn);
#pragma unroll
      for (int j = 0; j < 16; ++j) do_step(xcur[j], tc + j);
#pragma unroll
      for (int j = 0; j < 16; ++j) xcur[j] = xnxt[j];
    }
    for (int t = Tmain; t < T; ++t) do_step(x[rowoff + t], t);  // tail
  };

  if (base == 0) body(BoolT<true>{});   // block-uniform split: only this tile
  else           body(BoolT<false>{});  // pays for the batch-0 projection

  // ---- final hidden state: out[T + b*HID + h] (identity row mapping)
  float* hout = out + T;
#pragma unroll
  for (int v = 0; v < 8; ++v) {
    int h = v + 8 * half;
    if (h < HID && batchRow < B)
      hout[(size_t)batchRow * HID + h] = hv[v];
  }
}

extern "C" void kernel_launch(void* const* d_in, const int* in_sizes, int n_in,
                              void* d_out, int out_size, void* d_ws, size_t ws_size,
                              hipStream_t stream) {
  const float* x      = (const float*)d_in[0];
  const float* hidden = (const float*)d_in[1];
  const float* W_ih   = (const float*)d_in[2];
  const float* W_hh   = (const float*)d_in[3];
  const float* b_ih   = (const float*)d_in[4];
  const float* b_hh   = (const float*)d_in[5];
  const float* W_out  = (const float*)d_in[6];
  const float* b_out  = (const float*)d_in[7];

  const int B = in_sizes[1] / HID;   // hidden is (1, B, HID)
  const int T = in_sizes[0] / B;     // x is (B, T, 1)
  const int tiles = (B + 15) / 16;   // one wave32 per 16-batch tile

  rnn_wmma_kernel<<<tiles, 32, 0, stream>>>(x, hidden, W_ih, W_hh, b_ih, b_hh,
                                            W_out, b_out, (float*)d_out, B, T);
}